// PhysicsLoss_790273982719
// MI455X (gfx1250) — compile-verified
//
#include <hip/hip_runtime.h>
#include <hip/hip_bf16.h>

// ---------------------------------------------------------------------------
// PhysicsLoss fused pipeline for gfx1250 (MI455X).
// Memory-bound (~0.35 GB total traffic @ 23.3 TB/s ~= 15-20us). f32 throughout.
// CDNA5 specifics: ds_add_f32 LDS-privatized segment sums, and wave-level
// all-reduce built on V_WMMA_F32_16X16X4_F32 (ones-matmul row-sum trick).
// ---------------------------------------------------------------------------

typedef float v2f __attribute__((ext_vector_type(2)));
typedef float v8f __attribute__((ext_vector_type(8)));

#define W_SEG 2048
#define HIST  65536
#define ALPHA 0.1f
#define BETA  0.1f
#define CAPACITY 1000.0f

// -------- workspace word layout (floats/uints share d_ws) ------------------
#define OFF_CNT   0                 // float[2048] segment count
#define OFF_P     (W_SEG)           // float[2048] sum p
#define OFF_RATE  (2*W_SEG)         // float[2048] sum p*rate
#define OFF_DW    (3*W_SEG)         // float[2048] sum p*d_obs
#define OFF_SNLL  (4*W_SEG)         // float  sum(nll*wy)
#define OFF_SWY   (4*W_SEG+1)       // float  sum(wy)
#define OFF_NVF   (4*W_SEG+2)       // float  valid count (exact: < 2^24)
#define OFF_FRAC  (4*W_SEG+3)       // float  quantile frac
#define OFF_REF   (4*W_SEG+4)       // float  ref_dobs
#define OFF_BINLO (4*W_SEG+5)       // uint
#define OFF_RLO   (4*W_SEG+6)       // uint
#define OFF_BINHI (4*W_SEG+7)       // uint
#define OFF_RHI   (4*W_SEG+8)       // uint
#define OFF_NINT  (4*W_SEG+9)       // uint
#define OFF_H1    8448              // uint[65536] top-16-bit histogram
#define OFF_H2A   (OFF_H1 + HIST)   // uint[65536] low-16 refinement (rank lo)
#define OFF_H2B   (OFF_H1 + 2*HIST) // uint[65536] low-16 refinement (rank hi)
#define ZERO_WORDS (OFF_H1 + 3*HIST)  // 205056 words = ~0.82 MB of d_ws

// Wave32 all-reduce via WMMA: A(16x4) holds v in VGPR0 (K0 lanes0-15 /
// K2 lanes16-31) and 0 in VGPR1; B = ones. D[m][n] = v[m] + v[m+16]
// broadcast over n. Summing the 8 D VGPRs gives per-half partial sums;
// one shfl_xor(16) completes the 32-lane sum in every lane.
// Requires EXEC == all ones at the call site (call outside divergence).
__device__ __forceinline__ float wave_allreduce_wmma(float v) {
    v2f a; a[0] = v;    a[1] = 0.0f;
    v2f b; b[0] = 1.0f; b[1] = 1.0f;
    v8f c = {};
    c = __builtin_amdgcn_wmma_f32_16x16x4_f32(false, a, false, b,
                                              (short)0, c, false, false);
    float s = c[0] + c[1] + c[2] + c[3] + c[4] + c[5] + c[6] + c[7];
    s += __shfl_xor(s, 16, 32);
    return s;
}

__global__ void k_zero(unsigned* __restrict__ u, int nwords) {
    int stride = gridDim.x * blockDim.x;
    for (int i = blockIdx.x * blockDim.x + threadIdx.x; i < nwords; i += stride)
        u[i] = 0u;
}

// Fused main pass: CE partials + softmax prob + LDS-privatized segment sums +
// top-16-bit histogram of valid d_obs bit patterns.
__global__ __launch_bounds__(512)
void k_main(const float* __restrict__ logits, const int* __restrict__ y,
            const unsigned char* __restrict__ mask,   // jax bool = 1 byte
            const float* __restrict__ x_raw, const int* __restrict__ widx,
            const float* __restrict__ cw,
            float* __restrict__ wf, unsigned* __restrict__ wu, int n)
{
    __shared__ float sCnt[W_SEG], sP[W_SEG], sR[W_SEG], sD[W_SEG];
    __shared__ float rA[16], rB[16], rC[16];
    for (int s = threadIdx.x; s < W_SEG; s += blockDim.x) {
        sCnt[s] = 0.0f; sP[s] = 0.0f; sR[s] = 0.0f; sD[s] = 0.0f;
    }
    __syncthreads();

    const float cw0 = cw[0], cw1 = cw[1];
    float accN = 0.0f, accW = 0.0f, accC = 0.0f;
    const int stride = gridDim.x * blockDim.x;
    for (int i = blockIdx.x * blockDim.x + threadIdx.x; i < n; i += stride) {
        float2 lg = reinterpret_cast<const float2*>(logits)[i];
        int  yy = y[i];
        bool mk = mask[i] != 0;
        float l0 = lg.x, l1 = lg.y;
        float mmax = fmaxf(l0, l1);
        float lse  = mmax + __logf(__expf(l0 - mmax) + __expf(l1 - mmax));
        float nll  = lse - (yy ? l1 : l0);
        float w    = (yy ? cw1 : cw0) * (mk ? 1.0f : 0.0f);
        accN += nll * w;
        accW += w;

        int wi = widx[i];
        if (mk && wi >= 0) {
            float p = 1.0f / (1.0f + __expf(l0 - l1));  // softmax[:,1]
            // columns 2 (d_obs) and 3 (rate): 8B-aligned float2 load
            float2 xr = *reinterpret_cast<const float2*>(x_raw + (size_t)i * 8 + 2);
            float dob = fmaxf(xr.x, 0.0f);
            float fr  = fmaxf(xr.y, 0.0f);
            int s = wi & (W_SEG - 1);
            atomicAdd(&sCnt[s], 1.0f);       // ds_add_f32
            atomicAdd(&sP[s],   p);
            atomicAdd(&sR[s],   p * fr);
            atomicAdd(&sD[s],   p * dob);
            accC += 1.0f;
            atomicAdd(&wu[OFF_H1 + (__float_as_uint(dob) >> 16)], 1u);
        }
    }
    __syncthreads();

    for (int s = threadIdx.x; s < W_SEG; s += blockDim.x) {
        atomicAdd(&wf[OFF_CNT  + s], sCnt[s]);
        atomicAdd(&wf[OFF_P    + s], sP[s]);
        atomicAdd(&wf[OFF_RATE + s], sR[s]);
        atomicAdd(&wf[OFF_DW   + s], sD[s]);
    }

    // scalar reductions on the matrix pipe (EXEC all ones here)
    float wN = wave_allreduce_wmma(accN);
    float wW = wave_allreduce_wmma(accW);
    float wC = wave_allreduce_wmma(accC);
    int wave = threadIdx.x >> 5;
    if ((threadIdx.x & 31) == 0) { rA[wave] = wN; rB[wave] = wW; rC[wave] = wC; }
    __syncthreads();
    if (threadIdx.x == 0) {
        float tN = 0.0f, tW = 0.0f, tC = 0.0f;
        int nw = blockDim.x >> 5;
        for (int k = 0; k < nw; ++k) { tN += rA[k]; tW += rB[k]; tC += rC[k]; }
        atomicAdd(&wf[OFF_SNLL], tN);
        atomicAdd(&wf[OFF_SWY],  tW);
        atomicAdd(&wf[OFF_NVF],  tC);
    }
}

// Derive interpolation ranks and locate their top-16-bit bins.
__global__ void k_bins(float* __restrict__ wf, unsigned* __restrict__ wu) {
    int n = (int)(wf[OFF_NVF] + 0.5f);
    wu[OFF_NINT] = (unsigned)(n > 0 ? n : 0);
    if (n <= 0) {
        wf[OFF_FRAC] = 0.0f;
        wu[OFF_BINLO] = 0; wu[OFF_RLO] = 0; wu[OFF_BINHI] = 0; wu[OFF_RHI] = 0;
        return;
    }
    float pos = 0.75f * (float)(n - 1);     // exact: pos < 2^22
    if (pos < 0.0f) pos = 0.0f;
    unsigned lo = (unsigned)floorf(pos);
    unsigned hi = (unsigned)ceilf(pos);
    wf[OFF_FRAC] = pos - (float)lo;
    unsigned cum = 0; int fl = 0;
    for (unsigned b = 0; b < HIST; ++b) {
        unsigned c = wu[OFF_H1 + b];
        if (!fl && cum + c > lo) { wu[OFF_BINLO] = b; wu[OFF_RLO] = lo - cum; fl = 1; }
        if (cum + c > hi)        { wu[OFF_BINHI] = b; wu[OFF_RHI] = hi - cum; break; }
        cum += c;
    }
}

// Refinement scan: low-16-bit histograms restricted to the candidate bins.
__global__ __launch_bounds__(256)
void k_pass2(const unsigned char* __restrict__ mask, const int* __restrict__ widx,
             const float* __restrict__ x_raw, unsigned* __restrict__ wu, int n)
{
    if (wu[OFF_NINT] == 0) return;
    unsigned binLo = wu[OFF_BINLO], binHi = wu[OFF_BINHI];
    const int stride = gridDim.x * blockDim.x;
    for (int i = blockIdx.x * blockDim.x + threadIdx.x; i < n; i += stride) {
        bool mk = mask[i] != 0;
        int wi = widx[i];
        if (!(mk && wi >= 0)) continue;
        float dob = fmaxf(x_raw[(size_t)i * 8 + 2], 0.0f);
        unsigned bits = __float_as_uint(dob);
        unsigned top = bits >> 16, low = bits & 0xFFFFu;
        if (top == binLo) atomicAdd(&wu[OFF_H2A + low], 1u);
        if (top == binHi) atomicAdd(&wu[OFF_H2B + low], 1u);
    }
}

// Final selection + interpolation -> ref_dobs.
__global__ void k_select(float* __restrict__ wf, unsigned* __restrict__ wu) {
    if (wu[OFF_NINT] == 0) { wf[OFF_REF] = 1.0f; return; }
    unsigned binLo = wu[OFF_BINLO], rLo = wu[OFF_RLO];
    unsigned binHi = wu[OFF_BINHI], rHi = wu[OFF_RHI];
    unsigned cum = 0, lowLo = 0, lowHi = 0;
    for (unsigned b = 0; b < HIST; ++b) { cum += wu[OFF_H2A + b]; if (cum > rLo) { lowLo = b; break; } }
    cum = 0;
    for (unsigned b = 0; b < HIST; ++b) { cum += wu[OFF_H2B + b]; if (cum > rHi) { lowHi = b; break; } }
    float vLo = __uint_as_float((binLo << 16) | lowLo);
    float vHi = __uint_as_float((binHi << 16) | lowHi);
    float frac = wf[OFF_FRAC];
    float q = vLo * (1.0f - frac) + vHi * frac;
    wf[OFF_REF] = fmaxf(q, 1e-6f);
}

// Per-window finalize + loss assembly. 1024 threads x 2 windows each.
__global__ __launch_bounds__(1024)
void k_final(float* __restrict__ wf, float* __restrict__ out) {
    __shared__ float rF[32], rL[32], rK[32];
    const float ref = wf[OFF_REF];
    float fAcc = 0.0f, lAcc = 0.0f, kAcc = 0.0f;
    for (int s = threadIdx.x; s < W_SEG; s += blockDim.x) {
        float cnt = wf[OFF_CNT + s], ps = wf[OFF_P + s];
        float ar  = wf[OFF_RATE + s], dw = wf[OFF_DW + s];
        float d_mean = dw / (ps + 1e-6f);
        float rr   = ar / (CAPACITY + 1e-6f);
        float bu   = fmaxf(rr - 1.0f, 0.0f);
        float flow = bu * bu;
        float rho  = fminf(fmaxf(rr, 0.0f), 0.99f);
        float dth  = 1.0f / (1.0f - rho + 1e-6f);
        float lat  = fmaxf(dth - d_mean / ref, 0.0f);
        float keep = (cnt >= 2.0f && ps >= 1e-6f) ? 1.0f : 0.0f;
        fAcc += flow * keep; lAcc += lat * keep; kAcc += keep;
    }
    float wF = wave_allreduce_wmma(fAcc);
    float wL = wave_allreduce_wmma(lAcc);
    float wK = wave_allreduce_wmma(kAcc);
    int wave = threadIdx.x >> 5;
    if ((threadIdx.x & 31) == 0) { rF[wave] = wF; rL[wave] = wL; rK[wave] = wK; }
    __syncthreads();
    if (threadIdx.x == 0) {
        float f = 0.0f, l = 0.0f, k = 0.0f;
        int nw = blockDim.x >> 5;
        for (int i = 0; i < nw; ++i) { f += rF[i]; l += rL[i]; k += rK[i]; }
        float l_data = wf[OFF_SNLL] / wf[OFF_SWY];
        float denom  = fmaxf(k, 1.0f);
        float l_flow = (k > 0.0f) ? f / denom : 0.0f;
        float l_lat  = (k > 0.0f) ? l / denom : 0.0f;
        out[0] = l_data + ALPHA * l_flow + BETA * l_lat;
        out[1] = l_data;
        out[2] = l_flow;
        out[3] = l_lat;
    }
}

extern "C" void kernel_launch(void* const* d_in, const int* in_sizes, int n_in,
                              void* d_out, int out_size, void* d_ws, size_t ws_size,
                              hipStream_t stream)
{
    const float*         logits = (const float*)d_in[0];
    const int*           y      = (const int*)d_in[1];
    const unsigned char* mask   = (const unsigned char*)d_in[2];  // jax bool: 1B
    const float*         x_raw  = (const float*)d_in[3];
    const int*           widx   = (const int*)d_in[4];
    const float*         cw     = (const float*)d_in[5];
    float*    out = (float*)d_out;
    float*    wf  = (float*)d_ws;
    unsigned* wu  = (unsigned*)d_ws;   // needs ZERO_WORDS*4 ~= 0.82 MB of d_ws
    const int n = in_sizes[1];         // N rows (length of y)

    k_zero  <<<256, 256,  0, stream>>>(wu, ZERO_WORDS);
    k_main  <<<256, 512,  0, stream>>>(logits, y, mask, x_raw, widx, cw, wf, wu, n);
    k_bins  <<<1,   1,    0, stream>>>(wf, wu);
    k_pass2 <<<512, 256,  0, stream>>>(mask, widx, x_raw, wu, n);
    k_select<<<1,   1,    0, stream>>>(wf, wu);
    k_final <<<1,   1024, 0, stream>>>(wf, out);
}